// BlockDiagonal_52570399703165
// MI455X (gfx1250) — compile-verified
//
#include <hip/hip_runtime.h>

// Block-diagonal matmul: x[8192,4096] fp32, W[32,128,128] fp32, bias[4096] fp32.
// Memory-bound: ~270MB traffic -> ~11.6us at 23.3 TB/s. fp32 WMMA (16x16x4)
// keeps reference precision; matrix pipes have large slack.
//
// Per workgroup (256 thr = 8 waves): one (row_tile=128, kblock) pair.
//  - W[k] staged in LDS in paired-K layout W2[k/2][n][2], row stride 288 floats
//    (288%64==32 -> the two lane-halves of a B-fragment ds_load_b64 hit
//    disjoint 32-bank halves; all 64 banks used exactly once).
//  - x tile staged in LDS, row stride 132 floats (A-fragment b64 reads cover
//    all 64 banks exactly once). x loads + out stores are non-temporal so the
//    streaming traffic doesn't evict the L2-resident weights (2MB total).
//  - Each wave: 16 rows x 128 cols, 32 K-steps x 8 WMMAs, v8f accumulators.

#define SIZE_N 4096
#define NRB 32
#define BLK 128
#define WPAIR 288       // floats per paired-K LDS row (256 data + 32 pad)
#define XS 132          // LDS row stride (floats) for x tile
#define ROWS_PER_WG 128

typedef __attribute__((ext_vector_type(2))) float v2f;
typedef __attribute__((ext_vector_type(4))) float v4f;
typedef __attribute__((ext_vector_type(8))) float v8f;

__global__ __launch_bounds__(256)
void block_diag_wmma_f32(const float* __restrict__ x,
                         const float* __restrict__ w,
                         const float* __restrict__ bias,
                         float* __restrict__ out) {
  extern __shared__ float smem[];
  float* wsm = smem;                       // 64 * WPAIR floats (paired-K W)
  float* xsm = smem + 64 * WPAIR;          // 128 * XS floats

  const int kblk = blockIdx.x;             // 0..31
  const int rt   = blockIdx.y;             // row tile
  const int row0 = rt * ROWS_PER_WG;
  const int tid  = threadIdx.x;
  const int lane = tid & 31;
  const int wv   = tid >> 5;               // wave id 0..7
  const int lh   = lane & 15;
  const int hi   = lane >> 4;              // lane half: 0 or 1

  // ---- stage W[kblk] (128x128) into LDS in paired-K layout ----
  // W2[(k>>1)*WPAIR + 2*n + (k&1)] = W[k][n]
  const float* wg = w + (size_t)kblk * BLK * BLK;
  #pragma unroll
  for (int i = 0; i < 16; ++i) {
    const int idx = tid + i * 256;         // float4 index 0..4095
    const int k = idx >> 5, n0 = (idx & 31) * 4;
    const v4f v = *(const v4f*)(wg + (size_t)k * BLK + n0);
    float* dst = wsm + (k >> 1) * WPAIR + (k & 1);
    dst[(n0 + 0) * 2] = v.x;
    dst[(n0 + 1) * 2] = v.y;
    dst[(n0 + 2) * 2] = v.z;
    dst[(n0 + 3) * 2] = v.w;
  }

  // ---- stage x tile (128 rows x 128 cols) into LDS, non-temporal reads ----
  const float* xg = x + (size_t)row0 * SIZE_N + (size_t)kblk * BLK;
  #pragma unroll
  for (int i = 0; i < 16; ++i) {
    const int idx = tid + i * 256;
    const int r = idx >> 5, c4 = idx & 31;
    const v4f v = __builtin_nontemporal_load(
        (const v4f*)(xg + (size_t)r * SIZE_N + c4 * 4));
    *(v4f*)(xsm + r * XS + c4 * 4) = v;
  }

  __syncthreads();

  // ---- compute: wave wv owns rows [wv*16, wv*16+16), all 128 cols ----
  v8f acc[8] = {};

  const float* xrow = xsm + (wv * 16 + lh) * XS;  // A row (M = lh, both halves)
  const int koff = hi * 2;                        // lanes 16-31 hold K=+2,+3

  #pragma unroll 2
  for (int kk = 0; kk < BLK; kk += 4) {
    const v2f a = *(const v2f*)(xrow + kk + koff);          // K = kk+koff, +1
    // paired-K row: (kk+koff)/2 == kk/2 + hi; lane gives N = lh
    const float* wrow = wsm + ((kk >> 1) + hi) * WPAIR + lh * 2;
    #pragma unroll
    for (int nt = 0; nt < 8; ++nt) {
      const v2f b = *(const v2f*)(wrow + nt * 32);          // one ds_load_b64
      acc[nt] = __builtin_amdgcn_wmma_f32_16x16x4_f32(
          /*neg_a=*/false, a, /*neg_b=*/false, b,
          /*c_mod=*/(short)0, acc[nt], /*reuse_a=*/false, /*reuse_b=*/false);
    }
  }

  // ---- store with bias (non-temporal): VGPR r -> row (r + hi*8), N = lh ----
  const int cbase = kblk * BLK;
  float* orow = out + (size_t)(row0 + wv * 16 + hi * 8) * SIZE_N + cbase;
  #pragma unroll
  for (int nt = 0; nt < 8; ++nt) {
    const int col = nt * 16 + lh;
    const float bv = bias[cbase + col];
    #pragma unroll
    for (int r = 0; r < 8; ++r) {
      __builtin_nontemporal_store(acc[nt][r] + bv,
                                  orow + (size_t)r * SIZE_N + col);
    }
  }
}

extern "C" void kernel_launch(void* const* d_in, const int* in_sizes, int n_in,
                              void* d_out, int out_size, void* d_ws, size_t ws_size,
                              hipStream_t stream) {
  const float* x    = (const float*)d_in[0];   // [4,2048,4096] fp32
  const float* w    = (const float*)d_in[1];   // [32,128,128]  fp32
  const float* bias = (const float*)d_in[2];   // [4096]        fp32
  float* out        = (float*)d_out;

  const int nrows = in_sizes[0] / SIZE_N;      // 8192
  dim3 grid(NRB, nrows / ROWS_PER_WG);         // 32 x 64
  dim3 blk(256);
  const size_t shmem = (size_t)(64 * WPAIR + BLK * XS) * sizeof(float); // 141312 B

  block_diag_wmma_f32<<<grid, blk, shmem, stream>>>(x, w, bias, out);
}